// NT_Xent_76811195121782
// MI455X (gfx1250) — compile-verified
//
#include <hip/hip_runtime.h>
#include <hip/hip_bf16.h>

// ---------------------------------------------------------------------------
// NT-Xent (SimCLR) loss, B=2048, D=256, N=4096, T=0.5, eps=1e-8.
// fp32 WMMA (v_wmma_f32_16x16x4_f32) GEMM of zn @ zn^T with a streaming
// diagonal-masked sum-of-exp per row; |sim| <= 1/T = 2 so fp32 logsumexp
// needs no running max.
//
// The normalized matrix is stored PAIR-SWIZZLED per row:
//   znS[r][2k + j]       = zn[r][4k + j]      (j=0,1)  -> WMMA lanes 0..15
//   znS[r][128 + 2k + j] = zn[r][4k + 2 + j]  (j=0,1)  -> WMMA lanes 16..31
// so each lane's A/B fragment stream is contiguous: one global_load_b128
// feeds two 16x16x4 WMMAs.
// ---------------------------------------------------------------------------

typedef __attribute__((ext_vector_type(2))) float v2f;
typedef __attribute__((ext_vector_type(8))) float v8f;

#define NTX_B 2048
#define NTX_N 4096
#define NTX_D 256
#define NTX_NK (NTX_D / 4)       // 64 K-chunks of 4
#define NTX_NG (NTX_NK / 2)      // 32 float4 groups (2 K-chunks each)

// ---------------------------------------------------------------------------
// Kernel 1: row-normalize z = [z_i; z_j], fold 1/sqrt(T), store pair-swizzled.
// One wave per row.
// ---------------------------------------------------------------------------
__global__ __launch_bounds__(256) void ntxent_normalize(const float* __restrict__ zi,
                                                        const float* __restrict__ zj,
                                                        float* __restrict__ znS) {
    const int wave = threadIdx.x >> 5;
    const int lane = threadIdx.x & 31;
    const int row  = blockIdx.x * 8 + wave;          // 512 blocks * 8 waves = 4096 rows
    const float* src = (row < NTX_B) ? (zi + (size_t)row * NTX_D)
                                     : (zj + (size_t)(row - NTX_B) * NTX_D);
    // 256 floats / 32 lanes = 8 floats per lane as two coalesced float4s:
    //   v0 = elements 4*lane   .. 4*lane+3   (K-chunk k = lane)
    //   v1 = elements 128+4*lane .. +3       (K-chunk k = 32 + lane)
    float4 v0 = ((const float4*)src)[lane];
    float4 v1 = ((const float4*)src)[lane + 32];
    float ss = v0.x * v0.x + v0.y * v0.y + v0.z * v0.z + v0.w * v0.w
             + v1.x * v1.x + v1.y * v1.y + v1.z * v1.z + v1.w * v1.w;
    #pragma unroll
    for (int m = 1; m < 32; m <<= 1) ss += __shfl_xor(ss, m);
    const float norm  = sqrtf(ss);
    const float scale = 1.41421356237309515f /* sqrt(1/T) */ / fmaxf(norm, 1e-8f);

    float* dst = znS + (size_t)row * NTX_D;
    v2f lo0 = { v0.x * scale, v0.y * scale };   // chunk lane,    half 0
    v2f hi0 = { v0.z * scale, v0.w * scale };   // chunk lane,    half 1
    v2f lo1 = { v1.x * scale, v1.y * scale };   // chunk 32+lane, half 0
    v2f hi1 = { v1.z * scale, v1.w * scale };   // chunk 32+lane, half 1
    *(v2f*)(dst + 2 * lane)            = lo0;
    *(v2f*)(dst + 128 + 2 * lane)      = hi0;
    *(v2f*)(dst + 64 + 2 * lane)       = lo1;
    *(v2f*)(dst + 128 + 64 + 2 * lane) = hi1;
}

// ---------------------------------------------------------------------------
// Kernel 2: per-16-row stripe, streaming masked logsumexp over all 4096 cols.
// Block = 256 threads = 8 waves; wave w handles column tiles w, w+8, ...
// A fragments for all 64 K-chunks are preloaded into 128 VGPRs per lane.
// B is loaded as float4 (2 K-chunks per global_load_b128) with a 1-group
// prefetch so loads overlap the WMMA chain.
// ---------------------------------------------------------------------------
__global__ __launch_bounds__(256) void ntxent_main(const float* __restrict__ znS,
                                                   float* __restrict__ partials) {
    __shared__ float wsum[8][16];
    __shared__ float wpos[8][16];

    const int wave    = threadIdx.x >> 5;
    const int lane    = threadIdx.x & 31;
    const int half    = lane >> 4;     // 0: K pairs {4k,4k+1}, rows v; 1: {4k+2,4k+3}, rows v+8
    const int l16     = lane & 15;
    const int rowBase = blockIdx.x * 16;

    // Preload this stripe's A fragments (contiguous per lane after swizzle).
    v2f a[NTX_NK];
    {
        const float* aptr = znS + (size_t)(rowBase + l16) * NTX_D + 128 * half;
        #pragma unroll
        for (int g = 0; g < NTX_NG; ++g) {
            float4 t = *(const float4*)(aptr + 4 * g);
            a[2 * g + 0] = v2f{ t.x, t.y };
            a[2 * g + 1] = v2f{ t.z, t.w };
        }
    }

    float accS[8];   // per-row partial sum of exp (this lane's columns only)
    float accP[8];   // per-row positive-pair logit (at most one lane nonzero)
    #pragma unroll
    for (int v = 0; v < 8; ++v) { accS[v] = 0.0f; accP[v] = 0.0f; }

    for (int ct = wave; ct < NTX_N / 16; ct += 8) {
        const int colBase = ct * 16;
        const float* bptr = znS + (size_t)(colBase + l16) * NTX_D + 128 * half;

        float4 bnext = *(const float4*)(bptr);      // prefetch group 0
        v8f d = {};                                 // C starts at zero
        #pragma unroll
        for (int g = 0; g < NTX_NG; ++g) {
            const float4 bcur = bnext;
            if (g + 1 < NTX_NG)
                bnext = *(const float4*)(bptr + 4 * (g + 1));   // 1-ahead prefetch
            v2f b0 = { bcur.x, bcur.y };
            v2f b1 = { bcur.z, bcur.w };
            d = __builtin_amdgcn_wmma_f32_16x16x4_f32(false, a[2 * g + 0], false, b0,
                                                      (short)0, d, false, false);
            d = __builtin_amdgcn_wmma_f32_16x16x4_f32(false, a[2 * g + 1], false, b1,
                                                      (short)0, d, false, false);
        }

        // Lane holds column colBase+l16 for rows rowBase + v + 8*half.
        const int col = colBase + l16;
        #pragma unroll
        for (int v = 0; v < 8; ++v) {
            const int row = rowBase + v + 8 * half;
            const float s = d[v];
            accS[v] += (row == col) ? 0.0f : __expf(s);   // mask diagonal
            const int posCol = (row < NTX_B) ? (row + NTX_B) : (row - NTX_B);
            if (col == posCol) accP[v] += s;              // positive pair logit
        }
    }

    // Reduce over the 16 columns each half-wave owns (xor masks stay in-half).
    #pragma unroll
    for (int v = 0; v < 8; ++v) {
        float sS = accS[v], sP = accP[v];
        #pragma unroll
        for (int m = 1; m < 16; m <<= 1) {
            sS += __shfl_xor(sS, m);
            sP += __shfl_xor(sP, m);
        }
        if (l16 == 0) {
            wsum[wave][v + 8 * half] = sS;
            wpos[wave][v + 8 * half] = sP;
        }
    }
    __syncthreads();

    // Deterministic cross-wave combine + per-row lse - pos, then block sum.
    float val = 0.0f;
    if (threadIdx.x < 16) {
        float S = 0.0f, P = 0.0f;
        #pragma unroll
        for (int w = 0; w < 8; ++w) { S += wsum[w][threadIdx.x]; P += wpos[w][threadIdx.x]; }
        val = __logf(S) - P;
    }
    if (wave == 0) {
        #pragma unroll
        for (int m = 1; m < 16; m <<= 1) val += __shfl_xor(val, m);
        if (lane == 0) partials[blockIdx.x] = val;
    }
}

// ---------------------------------------------------------------------------
// Kernel 3: deterministic reduction of the 256 block partials -> loss / N.
// ---------------------------------------------------------------------------
__global__ __launch_bounds__(256) void ntxent_finish(const float* __restrict__ partials,
                                                     float* __restrict__ out) {
    __shared__ float red[8];
    const int t = threadIdx.x;
    float v = partials[t];
    #pragma unroll
    for (int m = 1; m < 32; m <<= 1) v += __shfl_xor(v, m);
    if ((t & 31) == 0) red[t >> 5] = v;
    __syncthreads();
    if (t < 8) {
        float s = red[t];
        #pragma unroll
        for (int m = 1; m < 8; m <<= 1) s += __shfl_xor(s, m);
        if (t == 0) out[0] = s * (1.0f / (float)NTX_N);
    }
}

// ---------------------------------------------------------------------------
extern "C" void kernel_launch(void* const* d_in, const int* in_sizes, int n_in,
                              void* d_out, int out_size, void* d_ws, size_t ws_size,
                              hipStream_t stream) {
    const float* zi = (const float*)d_in[0];
    const float* zj = (const float*)d_in[1];

    float* znS      = (float*)d_ws;                                   // 4 MB (swizzled)
    float* partials = (float*)((char*)d_ws + (size_t)NTX_N * NTX_D * sizeof(float));

    ntxent_normalize<<<NTX_N / 8, 256, 0, stream>>>(zi, zj, znS);
    ntxent_main<<<NTX_N / 16, 256, 0, stream>>>(znS, partials);
    ntxent_finish<<<1, 256, 0, stream>>>(partials, (float*)d_out);
}